// WaveSource_13838384627862
// MI455X (gfx1250) — compile-verified
//
#include <hip/hip_runtime.h>

typedef __attribute__((ext_vector_type(2))) float v2f;
typedef __attribute__((ext_vector_type(4))) float v4f;
typedef __attribute__((ext_vector_type(8))) float v8f;

#define MDIM 4096
#define KDIM 4096
#define NDIM 4096
#define SDIM 64
#define DT2  1.0e-6f

// ---------------- Kernel 1: out = Y (128 MB of traffic, pure streaming) ----------
__global__ __launch_bounds__(256) void ws_copy_kernel(const float* __restrict__ Y,
                                                      float* __restrict__ out, int n4) {
  int i = blockIdx.x * 256 + threadIdx.x;
  const int stride = gridDim.x * 256;
  const v4f* __restrict__ src = (const v4f*)Y;
  v4f* __restrict__ dst = (v4f*)out;
  for (; i < n4; i += stride) dst[i] = src[i];
}

// ---------------- Kernel 2: P = deno @ X via v_wmma_f32_16x16x4_f32, then
//                  out[:, x] = Y[:, x] + dt^2 * P  -----------------------------
// Block r (256 thr = 8 waves): rows [16r, 16r+16). Wave w reduces K in
// [512w, 512(w+1)) over all 4 N-tiles (N = 64), partial sums combined with
// ds_add_f32 in LDS, then scattered to the x columns.
__global__ __launch_bounds__(256) void ws_gemm_scatter_kernel(
    const float* __restrict__ Y, const float* __restrict__ X,
    const float* __restrict__ deno, const int* __restrict__ xidx,
    float* __restrict__ out) {
  __shared__ __align__(16) float acc_s[16 * 64];

  const int tid  = threadIdx.x;
  const int wid  = tid >> 5;        // wave 0..7 -> K slice
  const int lane = tid & 31;
  const int l16  = lane & 15;
  const int lh   = lane >> 4;       // 0: K slots 0,1 | 1: K slots 2,3
  const int r    = blockIdx.x;      // row tile

  // zero the shared 16x64 accumulator tile (4 KB)
  {
    v4f z = {0.f, 0.f, 0.f, 0.f};
    ((v4f*)acc_s)[tid] = z;
  }
  __syncthreads();

  // Lane's A base: row (16r + l16) of deno, plus slot-half offset.
  // Lanes 0-15 stream deno[m, k0..k0+3], lanes 16-31 deno[m, k0+4..k0+7]:
  // one b128 per lane feeds TWO wmma steps (slot remapping of K).
  const float* __restrict__ arow =
      deno + (size_t)(16 * r + l16) * (size_t)KDIM + (size_t)(lh * 4);

  v8f acc[4];
#pragma unroll
  for (int t = 0; t < 4; ++t) {
    v8f z = {0.f, 0.f, 0.f, 0.f, 0.f, 0.f, 0.f, 0.f};
    acc[t] = z;
  }

  int k0 = 512 * wid;
#pragma unroll 2
  for (int it = 0; it < 64; ++it, k0 += 8) {
    const v4f a4 = *(const v4f*)(arow + k0);
    // stream-prefetch deno ~8 iterations (64 floats) ahead -> global_prefetch_b8
    __builtin_prefetch((const void*)(arow + k0 + 64), 0, 1);

    v2f a01 = {a4.x, a4.y};   // step 0: physical K {k0,k0+1} / {k0+4,k0+5}
    v2f a23 = {a4.z, a4.w};   // step 1: physical K {k0+2,k0+3} / {k0+6,k0+7}

    const int kb = k0 + lh * 4;                      // lane's B row base
    const float* __restrict__ xb = X + (size_t)kb * SDIM + l16;

#pragma unroll
    for (int t = 0; t < 4; ++t) {
      v2f b0 = {xb[16 * t], xb[16 * t + SDIM]};             // X[kb,n], X[kb+1,n]
      v2f b1 = {xb[16 * t + 2 * SDIM], xb[16 * t + 3 * SDIM]};  // X[kb+2,n], X[kb+3,n]
      acc[t] = __builtin_amdgcn_wmma_f32_16x16x4_f32(
          false, a01, false, b0, (short)0, acc[t], false, false);
      acc[t] = __builtin_amdgcn_wmma_f32_16x16x4_f32(
          false, a23, false, b1, (short)0, acc[t], false, false);
    }
  }

  // Combine the 8 K-slices: ds_add_f32 into the shared tile.
  // C/D layout: VGPR v of tile t holds row M = v + 8*lh, col n = 16t + l16.
#pragma unroll
  for (int t = 0; t < 4; ++t) {
#pragma unroll
    for (int v = 0; v < 8; ++v) {
      const int row = v + 8 * lh;
      const int col = 16 * t + l16;
      atomicAdd(&acc_s[row * 64 + col], acc[t][v]);
    }
  }
  __syncthreads();

  // Scatter: each thread owns 4 consecutive tile elements.
  const int i0   = tid * 4;
  const int mrow = i0 >> 6;        // 0..15
  const int c0   = i0 & 63;        // 0..60 step 4
  const v4f p    = ((const v4f*)acc_s)[tid];
  const size_t rowbase = (size_t)(16 * r + mrow) * (size_t)NDIM;
#pragma unroll
  for (int j = 0; j < 4; ++j) {
    const int col = xidx[c0 + j];
    out[rowbase + col] = Y[rowbase + col] + DT2 * p[j];
  }
}

extern "C" void kernel_launch(void* const* d_in, const int* in_sizes, int n_in,
                              void* d_out, int out_size, void* d_ws, size_t ws_size,
                              hipStream_t stream) {
  const float* Y    = (const float*)d_in[0];
  const float* X    = (const float*)d_in[1];
  const float* deno = (const float*)d_in[2];
  const int*   xidx = (const int*)d_in[3];
  float* out = (float*)d_out;

  // 1) out = Y  (16M floats -> 4M float4)
  ws_copy_kernel<<<4096, 256, 0, stream>>>(Y, out, (MDIM * NDIM) / 4);

  // 2) WMMA GEMM + scatter-add into the x columns (runs after copy in-stream)
  ws_gemm_scatter_kernel<<<MDIM / 16, 256, 0, stream>>>(Y, X, deno, xidx, out);

  (void)in_sizes; (void)n_in; (void)out_size; (void)d_ws; (void)ws_size;
}